// PositionalEncoding_28587302322277
// MI455X (gfx1250) — compile-verified
//
#include <hip/hip_runtime.h>
#include <hip/hip_bf16.h>
#include <math.h>
#include <stdint.h>

// Output row: [coords(3) | enc(64) | colours(3)] = 70 f32 = 280 B.
// Tile: 64 rows x 70 cols = 17920 B = 1120 x 16B chunks (16B-aligned per block).
#define OUT_COLS        70
#define ROWS_PER_BLOCK  64
#define THREADS         256
#define TILE_FLOATS     (ROWS_PER_BLOCK * OUT_COLS)   // 4480
#define TILE_CHUNKS     ((TILE_FLOATS * 4) / 16)      // 1120

// ln(10000); reference computes power(10000, 2j/64) ~ exp((j/32) * ln(10000))
#define LN_10000 9.2103403719761836f

__global__ __launch_bounds__(THREADS)
void PositionalEncoding_28587302322277_kernel(const float* __restrict__ in,
                                              float* __restrict__ out, int n) {
    __shared__ float tile[TILE_FLOATS];
    const int tid     = threadIdx.x;
    const int rowBase = blockIdx.x * ROWS_PER_BLOCK;
    int rowsHere = n - rowBase;
    if (rowsHere > ROWS_PER_BLOCK) rowsHere = ROWS_PER_BLOCK;
    if (rowsHere <= 0) return;

    // ---- Phase 1a: coords + colours, one row per thread (no div/mod). ----
    // Lanes read 24B-apart rows (dense 6-line coverage per wave); LDS writes at
    // stride 70 floats -> banks 6*lane mod 64 -> 32 distinct banks, conflict-free.
    if (tid < rowsHere) {
        const float* rp = in + (size_t)(rowBase + tid) * 6;
        float*       tp = tile + tid * OUT_COLS;
        tp[0]  = rp[0];  tp[1]  = rp[1];  tp[2]  = rp[2];   // coords -> cols 0..2
        tp[67] = rp[3];  tp[68] = rp[4];  tp[69] = rp[5];   // colours -> cols 67..69
    }

    // ---- Phase 1b: positional encoding. lane = freq index j, wave = row group.
    // Mirror the reference rounding path: p = exp((j/32)*ln 1e4); pos = i / p.
    // One accurate sincosf (full range reduction; angles reach ~1e6) fills both
    // slots. LDS writes are stride-2 floats across lanes -> 32 distinct banks.
    const int   j  = tid & 31;
    const int   rG = tid >> 5;                       // wave id 0..7
    const float y  = (float)j * 0.03125f;            // 2j/64, exact in f32
    const float p  = expf(y * LN_10000);             // 10000^(2j/64)
    #pragma unroll
    for (int k = 0; k < 8; ++k) {
        int r = rG * 8 + k;                          // 0..63
        if (r < rowsHere) {
            float pos = (float)(rowBase + r) / p;
            float s, c;
            sincosf(pos, &s, &c);
            tile[r * OUT_COLS + 3 + 2 * j]     = s;
            tile[r * OUT_COLS + 3 + 2 * j + 1] = c;
        }
    }

    __syncthreads();

    // ---- Phase 2: stream tile -> global ----
    char* gbase = (char*)(out + (size_t)rowBase * OUT_COLS);
    if (rowsHere == ROWS_PER_BLOCK) {
        // CDNA5 async LDS->global 128-bit streaming stores (ASYNCcnt path).
        // Consecutive lanes write consecutive 16B chunks: fully coalesced.
        // NT hint: 280MB output > 192MB L2, written once, never re-read.
        for (int chunk = tid; chunk < TILE_CHUNKS; chunk += THREADS) {
            unsigned loff     = (unsigned)(chunk << 4);
            unsigned lds_addr = (unsigned)(uintptr_t)tile + loff;   // low 32b = LDS byte addr
            uint64_t gaddr    = (uint64_t)(uintptr_t)gbase + loff;
            asm volatile("global_store_async_from_lds_b128 %0, %1, off th:TH_STORE_NT"
                         :: "v"(gaddr), "v"(lds_addr)
                         : "memory");
        }
        asm volatile("s_wait_asynccnt 0x0" ::: "memory");
    } else {
        // Partial tail tile (not hit for N=1e6): plain coalesced f32 stores.
        for (int idx = tid; idx < rowsHere * OUT_COLS; idx += THREADS)
            out[(size_t)rowBase * OUT_COLS + idx] = tile[idx];
    }
}

extern "C" void kernel_launch(void* const* d_in, const int* in_sizes, int n_in,
                              void* d_out, int out_size, void* d_ws, size_t ws_size,
                              hipStream_t stream) {
    const float* in  = (const float*)d_in[0];
    float*       out = (float*)d_out;
    const int n = in_sizes[0] / 6;                       // 1,000,000 points
    const int blocks = (n + ROWS_PER_BLOCK - 1) / ROWS_PER_BLOCK;  // 15625 full tiles
    PositionalEncoding_28587302322277_kernel<<<blocks, THREADS, 0, stream>>>(in, out, n);
}